// QuantumLayer_23725399343174
// MI455X (gfx1250) — compile-verified
//
#include <hip/hip_runtime.h>
#include <hip/hip_bf16.h>

typedef __attribute__((ext_vector_type(2))) float v2f;
typedef __attribute__((ext_vector_type(8))) float v8f;

__device__ __forceinline__ float gsign(int k, int mask) {
    // (-1)^popcount(k & mask) as float
    return 1.0f - 2.0f * (float)(__popc(k & mask) & 1);
}

// out[b,i] = prod_{m in {i,i-2,...}} cos(x[b,m]), computed as
// OUT_tile = (PA @ GA) .* (PB @ GB) per 16-row tile via v_wmma_f32_16x16x4_f32.
__global__ __launch_bounds__(256) void qexp_kernel(const float* __restrict__ x,
                                                   float* __restrict__ out,
                                                   int B, int tiles) {
    const int lane = threadIdx.x & 31;
    const int tile = blockIdx.x * (blockDim.x >> 5) + (threadIdx.x >> 5);
    if (tile >= tiles) return;            // wave-uniform: EXEC stays all-ones
    const int  m  = lane & 15;
    const bool hi = lane >= 16;

    // ---- load this row's 10 inputs (rows are 40B -> 8B aligned; 5x b64 loads)
    int row  = tile * 16 + m;
    int lrow = row < B ? row : B - 1;     // safe clamp (B=4096 is exact anyway)
    const float* xr = x + (size_t)lrow * 10;

    float c2[10], s2[10];                 // cos^2(x/2), sin^2(x/2) via half-angle
#pragma unroll
    for (int w = 0; w < 5; ++w) {
        v2f t = *(const v2f*)(xr + 2 * w);
        float ca = __cosf(t.x);
        float cb = __cosf(t.y);
        c2[2*w]   = 0.5f * (1.0f + ca);  s2[2*w]   = 0.5f * (1.0f - ca);
        c2[2*w+1] = 0.5f * (1.0f + cb);  s2[2*w+1] = 0.5f * (1.0f - cb);
    }

    // ---- product trees: base[s] covers wires {1,2,3,4} (A) / {6,7,8,9} (B).
    // K-index bit j <-> wire j (half A) / wire 5+j (half B).
    // Lane-hi lanes supply K with bit1=1, folded in via f1.
    float baseA[8], baseB[8];
    {
        float f1 = hi ? s2[1] : c2[1];
        float t0 = f1 * c2[2], t1 = f1 * s2[2];
        float u0 = t0 * c2[3], u1 = t1 * c2[3], u2 = t0 * s2[3], u3 = t1 * s2[3];
        baseA[0] = u0 * c2[4]; baseA[1] = u1 * c2[4];
        baseA[2] = u2 * c2[4]; baseA[3] = u3 * c2[4];
        baseA[4] = u0 * s2[4]; baseA[5] = u1 * s2[4];
        baseA[6] = u2 * s2[4]; baseA[7] = u3 * s2[4];
    }
    {
        float f1 = hi ? s2[6] : c2[6];
        float t0 = f1 * c2[7], t1 = f1 * s2[7];
        float u0 = t0 * c2[8], u1 = t1 * c2[8], u2 = t0 * s2[8], u3 = t1 * s2[8];
        baseB[0] = u0 * c2[9]; baseB[1] = u1 * c2[9];
        baseB[2] = u2 * c2[9]; baseB[3] = u3 * c2[9];
        baseB[4] = u0 * s2[9]; baseB[5] = u1 * s2[9];
        baseB[6] = u2 * s2[9]; baseB[7] = u3 * s2[9];
    }

    // ---- per-column parity masks (column index = lane&15; cols >=10 are junk pads)
    // S_i = {i, i-2, ...}; half-A mask bit j <-> wire j, half-B bit j <-> wire 5+j.
    int mskA = (m & 1) ? ((m == 1) ? 2 : 10)
                       : ((m == 0) ? 1 : (m == 2) ? 5 : 21);
    int jb   = m - 5;
    int mskB = (jb < 0) ? 0 : (jb == 0) ? 1 : (jb == 1) ? 2
             : (jb == 2) ? 5 : (jb == 3) ? 10 : 21;

    // ---- two chained WMMA accumulations, K = 32 in 8 steps of 4
    v8f accA = {0.f,0.f,0.f,0.f,0.f,0.f,0.f,0.f};
    v8f accB = {0.f,0.f,0.f,0.f,0.f,0.f,0.f,0.f};
    const int koff = hi ? 2 : 0;
#pragma unroll
    for (int s = 0; s < 8; ++s) {
        int k = 4 * s + koff;
        v2f a, b;
        // A operand: PA[row, k] = base[s] * (bit0 ? sin^2 : cos^2)(wire0)
        a.x = baseA[s] * c2[0];  a.y = baseA[s] * s2[0];
        // B operand: GA[k, col] = (-1)^popc(k & mask)
        b.x = gsign(k, mskA);    b.y = gsign(k + 1, mskA);
        accA = __builtin_amdgcn_wmma_f32_16x16x4_f32(
                   false, a, false, b, (short)0, accA, false, false);

        a.x = baseB[s] * c2[5];  a.y = baseB[s] * s2[5];
        b.x = gsign(k, mskB);    b.y = gsign(k + 1, mskB);
        accB = __builtin_amdgcn_wmma_f32_16x16x4_f32(
                   false, a, false, b, (short)0, accB, false, false);
    }

    // ---- combine halves elementwise (same D layout) and store cols 0..9
    if (m < 10) {
        int rbase = tile * 16 + (hi ? 8 : 0);
#pragma unroll
        for (int v = 0; v < 8; ++v) {
            int r = rbase + v;
            if (r < B) out[(size_t)r * 10 + m] = accA[v] * accB[v];
        }
    }
}

extern "C" void kernel_launch(void* const* d_in, const int* in_sizes, int n_in,
                              void* d_out, int out_size, void* d_ws, size_t ws_size,
                              hipStream_t stream) {
    const float* x = (const float*)d_in[0];
    // d_in[1] (q_weights) provably does not affect the output: all its phases
    // are unit-modulus and cancel in |state|^2.
    float* out = (float*)d_out;

    int B     = in_sizes[0] / 10;         // 4096 rows of 10 features
    int tiles = (B + 15) / 16;            // 16 rows per wave tile
    const int wavesPerBlock = 8;          // 256 threads = 8 wave32
    int blocks = (tiles + wavesPerBlock - 1) / wavesPerBlock;

    qexp_kernel<<<blocks, 32 * wavesPerBlock, 0, stream>>>(x, out, B, tiles);
}